// ALSHLinear_56014963474549
// MI455X (gfx1250) — compile-verified
//
#include <hip/hip_runtime.h>

// ---------------- ALSH constants (match reference) ----------------
#define M_AUG      5
#define TABLE_SIZE 64
#define R_BIN      4.0f
#define U_SCALE    0.83f

typedef __attribute__((ext_vector_type(2))) float v2f;
typedef __attribute__((ext_vector_type(4))) float v4f;
typedef __attribute__((ext_vector_type(8))) float v8f;
typedef __attribute__((__vector_size__(16))) int v4i;   // matches builtin param type

// ---- CDNA5 async global->LDS copy (ASYNCcnt path), with safe fallbacks ----
__device__ __forceinline__ void async_copy_b128(void* lds_dst, const void* gsrc) {
#if defined(__gfx1250__) && __has_builtin(__builtin_amdgcn_global_load_async_to_lds_b128)
  __builtin_amdgcn_global_load_async_to_lds_b128(
      (__attribute__((address_space(1))) v4i*)(gsrc),
      (__attribute__((address_space(3))) v4i*)(lds_dst),
      0, 0);
#else
  v4f v = *(const v4f*)gsrc;
  *(v4f*)lds_dst = v;
#endif
}

__device__ __forceinline__ void async_wait_all() {
#if defined(__gfx1250__) && __has_builtin(__builtin_amdgcn_s_wait_asynccnt)
  __builtin_amdgcn_s_wait_asynccnt(0);
#elif defined(__gfx1250__) && __has_builtin(__builtin_amdgcn_global_load_async_to_lds_b128)
  asm volatile("s_wait_asynccnt 0x0" ::: "memory");
#endif
}

__device__ __forceinline__ int bucket_of(float dot) {
  // floor(dot / R) mod 64 with Python-style (non-negative) modulo; abs is a no-op.
  float h = floorf(dot / R_BIN);
  float m = fmodf(h, (float)TABLE_SIZE);
  if (m < 0.0f) m += (float)TABLE_SIZE;
  return (int)m;
}

// ---------------- 0: init scratch scalars ----------------
__global__ void alsh_init_kernel(unsigned int* maxbits, int* act_cnt) {
  if (threadIdx.x == 0) { *maxbits = 0u; *act_cnt = 0; }
}

// ---------------- 1: per-row ||W||^2 and W.a, global max ||W||^2 ----------------
// Streams W (268 MB) once with non-temporal loads: it does not fit L2 (192 MB)
// and only ~1/64 of rows are re-read later by the gather GEMM.
__global__ __launch_bounds__(256) void alsh_row_stats_kernel(
    const float* __restrict__ W, const float* __restrict__ a, int D,
    float* __restrict__ s2, float* __restrict__ sa, unsigned int* __restrict__ maxbits) {
  __shared__ float r1[256], r2[256];
  const int n = blockIdx.x;
  const v4f* row4 = (const v4f*)(W + (size_t)n * D);
  const v4f* a4   = (const v4f*)a;
  const int D4 = D >> 2;
  float acc2 = 0.f, acca = 0.f;
  for (int d = threadIdx.x; d < D4; d += 256) {
    v4f w = __builtin_nontemporal_load(row4 + d);
    v4f av = a4[d];
    acc2 += w[0]*w[0] + w[1]*w[1] + w[2]*w[2] + w[3]*w[3];
    acca += w[0]*av[0] + w[1]*av[1] + w[2]*av[2] + w[3]*av[3];
  }
  r1[threadIdx.x] = acc2; r2[threadIdx.x] = acca;
  __syncthreads();
  for (int s = 128; s > 0; s >>= 1) {
    if (threadIdx.x < s) {
      r1[threadIdx.x] += r1[threadIdx.x + s];
      r2[threadIdx.x] += r2[threadIdx.x + s];
    }
    __syncthreads();
  }
  if (threadIdx.x == 0) {
    s2[n] = r1[0];
    sa[n] = r2[0];
    atomicMax(maxbits, __float_as_uint(r1[0]));  // bits monotone for >=0 floats
  }
}

// ---------------- 2: column mean of x ----------------
__global__ void alsh_colmean_kernel(const float* __restrict__ x, int Bn, int D,
                                    float* __restrict__ ave) {
  int d = blockIdx.x * blockDim.x + threadIdx.x;
  if (d >= D) return;
  float s = 0.f;
  for (int b = 0; b < Bn; ++b) s += x[(size_t)b * D + d];
  ave[d] = s / (float)Bn;
}

// ---------------- 3: query bucket + ALSH scale ----------------
__global__ __launch_bounds__(256) void alsh_qhash_kernel(
    const float* __restrict__ ave, const float* __restrict__ a, int D,
    const unsigned int* __restrict__ maxbits, int* __restrict__ qb,
    float* __restrict__ scale_out) {
  __shared__ float r1[256], r2[256];
  float n2 = 0.f, da = 0.f;
  for (int d = threadIdx.x; d < D; d += 256) {
    float v = ave[d];
    n2 += v * v;
    da += v * a[d];
  }
  r1[threadIdx.x] = n2; r2[threadIdx.x] = da;
  __syncthreads();
  for (int s = 128; s > 0; s >>= 1) {
    if (threadIdx.x < s) {
      r1[threadIdx.x] += r1[threadIdx.x + s];
      r2[threadIdx.x] += r2[threadIdx.x + s];
    }
    __syncthreads();
  }
  if (threadIdx.x == 0) {
    float dot = r2[0] / sqrtf(r1[0]);         // q = ave/||ave||
    float tail = 0.f;
    for (int i = 0; i < M_AUG; ++i) tail += a[D + i];
    dot += 0.5f * tail;                        // Q-augmentation: m copies of 0.5
    *qb = bucket_of(dot);
    *scale_out = U_SCALE / sqrtf(__uint_as_float(*maxbits));
  }
}

// ---------------- 4: row buckets -> compact active index list ----------------
__global__ void alsh_mask_kernel(const float* __restrict__ s2, const float* __restrict__ sa,
                                 const float* __restrict__ a, int D, int N,
                                 const int* __restrict__ qb, const float* __restrict__ scale,
                                 int* __restrict__ act_idx, int* __restrict__ act_cnt) {
  int n = blockIdx.x * blockDim.x + threadIdx.x;
  if (n >= N) return;
  const float s  = *scale;
  const float n2 = s * s * s2[n];             // ||Ws_n||^2
  float dot = s * sa[n];                       // Ws_n . a[:D]
  float p = n2;                                // n2^(2^i) chain: n2, n2^2, n2^4, ...
  for (int i = 0; i < M_AUG; ++i) { dot += p * a[D + i]; p = p * p; }
  if (bucket_of(dot) == *qb) {
    int pos = atomicAdd(act_cnt, 1);
    act_idx[pos] = n;
  }
}

// ---------------- 5: zero the (mostly-sparse) output ----------------
// 268 MB streaming store: non-temporal, keep it out of L2.
__global__ void alsh_zero_kernel(v4f* __restrict__ out, size_t n4) {
  size_t i = (size_t)blockIdx.x * blockDim.x + threadIdx.x;
  size_t stride = (size_t)gridDim.x * blockDim.x;
  v4f z = {0.f, 0.f, 0.f, 0.f};
  for (; i < n4; i += stride) __builtin_nontemporal_store(z, out + i);
}

// ---------------- 6: WMMA GEMM over compacted active columns ----------------
// out[b, act_idx[c]] = sum_d x[b,d] * W[act_idx[c], d]
#define BM   128
#define BN   128
#define KB   16
#define LDSS 20   // 16 + 4 pad: rows 16B-aligned (async b128 dst) AND
                  // 20*r mod 64 distinct for r=0..15 -> conflict-free fragment reads

__global__ __launch_bounds__(256) void alsh_gemm_kernel(
    const float* __restrict__ x, const float* __restrict__ W,
    const int* __restrict__ act_idx, const int* __restrict__ act_cnt,
    float* __restrict__ out, int Bn, int D, int N) {
  __shared__ float As[BM * LDSS];
  __shared__ float Bs[BN * LDSS];
  __shared__ int   cidx[BN];

  const int count = *act_cnt;
  const int cbase = blockIdx.y * BN;
  if (cbase >= count) return;   // uniform exit: out already zeroed

  const int tid = threadIdx.x;
  if (tid < BN) {
    int c = cbase + tid;
    cidx[tid] = (c < count) ? act_idx[c] : -1;  // -1 => zero-padded column
  }
  __syncthreads();

  const int wave   = tid >> 5;           // 8 waves (wave32)
  const int lane   = tid & 31;
  const int waveM  = (wave >> 1) * 32;   // 4 waves along M
  const int waveN  = (wave & 1) * 64;    // 2 waves along N
  const int lm     = lane & 15;          // A/B row within 16-tile
  const int lh     = lane >> 4;          // half-wave selector
  const int lk     = lh << 1;            // K base {0,2} per ISA A/B layout
  const int blockM = blockIdx.x * BM;

  v8f acc[2][4];
#pragma unroll
  for (int mi = 0; mi < 2; ++mi)
#pragma unroll
    for (int ni = 0; ni < 4; ++ni)
#pragma unroll
      for (int i = 0; i < 8; ++i) acc[mi][ni][i] = 0.0f;

  for (int k0 = 0; k0 < D; k0 += KB) {
    // Stage A tile: BM x KB via async global->LDS (512 b128 copies, 2/thread)
#pragma unroll
    for (int i = 0; i < 2; ++i) {
      int idx = tid + i * 256;
      int r   = idx >> 2;            // KB/4 = 4 b128 per row
      int c4  = idx & 3;
      async_copy_b128(&As[r * LDSS + c4 * 4],
                      x + (size_t)(blockM + r) * D + k0 + c4 * 4);
    }
    // Stage B tile: gathered active W rows; zeros for padded columns
#pragma unroll
    for (int i = 0; i < 2; ++i) {
      int idx = tid + i * 256;
      int r   = idx >> 2;
      int c4  = idx & 3;
      int gi  = cidx[r];
      float* dst = &Bs[r * LDSS + c4 * 4];
      if (gi >= 0) {
        async_copy_b128(dst, W + (size_t)gi * D + k0 + c4 * 4);
      } else {
        dst[0] = 0.f; dst[1] = 0.f; dst[2] = 0.f; dst[3] = 0.f;
      }
    }
    async_wait_all();
    __syncthreads();

#pragma unroll
    for (int kk = 0; kk < KB; kk += 4) {
      // A fragment: lane holds M=lm, K={lk, lk+1}   (ISA 7.12.2, 32-bit A 16x4)
      v2f af[2];
#pragma unroll
      for (int mi = 0; mi < 2; ++mi) {
        const float* p = &As[(waveM + mi * 16 + lm) * LDSS + kk + lk];
        af[mi][0] = p[0]; af[mi][1] = p[1];
      }
      // B fragment: lane holds N=lm, K={lk, lk+1}   (B stored row=N, col=K in LDS)
      v2f bf[4];
#pragma unroll
      for (int ni = 0; ni < 4; ++ni) {
        const float* p = &Bs[(waveN + ni * 16 + lm) * LDSS + kk + lk];
        bf[ni][0] = p[0]; bf[ni][1] = p[1];
      }
#pragma unroll
      for (int mi = 0; mi < 2; ++mi)
#pragma unroll
        for (int ni = 0; ni < 4; ++ni)
          acc[mi][ni] = __builtin_amdgcn_wmma_f32_16x16x4_f32(
              false, af[mi], false, bf[ni], (short)0, acc[mi][ni], false, false);
    }
    __syncthreads();
  }

  // Store: C/D layout lane -> (N = lane&15, M = vgpr + 8*(lane>>4)); out never re-read.
#pragma unroll
  for (int ni = 0; ni < 4; ++ni) {
    int ncol = cidx[waveN + ni * 16 + lm];
#pragma unroll
    for (int mi = 0; mi < 2; ++mi) {
      v8f c = acc[mi][ni];
      if (ncol >= 0) {
#pragma unroll
        for (int i = 0; i < 8; ++i) {
          int mrow = blockM + waveM + mi * 16 + i + 8 * lh;
          __builtin_nontemporal_store(c[i], &out[(size_t)mrow * N + ncol]);
        }
      }
    }
  }
}

// ---------------- host-side launch ----------------
extern "C" void kernel_launch(void* const* d_in, const int* in_sizes, int n_in,
                              void* d_out, int out_size, void* d_ws, size_t ws_size,
                              hipStream_t stream) {
  const float* x = (const float*)d_in[0];
  const float* W = (const float*)d_in[1];
  const float* a = (const float*)d_in[2];
  // d_in[3] = mode, unused (reference ignores it)

  const int D  = in_sizes[2] - M_AUG;      // 2048
  const int Bn = in_sizes[0] / D;          // 2048
  const int N  = in_sizes[1] / D;          // 32768
  float* out = (float*)d_out;

  // Scratch layout
  float* s2       = (float*)d_ws;              // N
  float* sa       = s2 + N;                    // N
  int*   act_idx  = (int*)(sa + N);            // N
  float* ave      = (float*)(act_idx + N);     // D
  unsigned int* maxbits = (unsigned int*)(ave + D);
  int*   act_cnt  = (int*)(maxbits + 1);
  int*   qb       = act_cnt + 1;
  float* scale    = (float*)(qb + 1);

  hipLaunchKernelGGL(alsh_init_kernel, dim3(1), dim3(64), 0, stream, maxbits, act_cnt);
  hipLaunchKernelGGL(alsh_row_stats_kernel, dim3(N), dim3(256), 0, stream,
                     W, a, D, s2, sa, maxbits);
  hipLaunchKernelGGL(alsh_colmean_kernel, dim3((D + 255) / 256), dim3(256), 0, stream,
                     x, Bn, D, ave);
  hipLaunchKernelGGL(alsh_qhash_kernel, dim3(1), dim3(256), 0, stream,
                     ave, a, D, maxbits, qb, scale);
  hipLaunchKernelGGL(alsh_mask_kernel, dim3((N + 255) / 256), dim3(256), 0, stream,
                     s2, sa, a, D, N, qb, scale, act_idx, act_cnt);

  size_t n4 = ((size_t)Bn * (size_t)N) / 4;
  hipLaunchKernelGGL(alsh_zero_kernel, dim3(4096), dim3(256), 0, stream,
                     (v4f*)out, n4);

  dim3 grid(Bn / BM, N / BN);   // worst case: all rows active; blocks past count exit
  hipLaunchKernelGGL(alsh_gemm_kernel, grid, dim3(256), 0, stream,
                     x, W, act_idx, act_cnt, out, Bn, D, N);
}